// GCN_63153199120909
// MI455X (gfx1250) — compile-verified
//
#include <hip/hip_runtime.h>
#include <math.h>

#define NF 512
#define NH 16
#define NC 40
#define KSTEPS (NF / 4)   // 128 WMMA K-steps of 4

typedef __attribute__((ext_vector_type(2))) float v2f;
typedef __attribute__((ext_vector_type(8))) float v8f;

// ---------------------------------------------------------------- utilities
__global__ void fill_f32(float* __restrict__ p, float v, int n) {
  int i = blockIdx.x * blockDim.x + threadIdx.x;
  if (i < n) p[i] = v;
}

// deg starts at 1.0 (self-loop), count incoming edges by dst
__global__ void deg_count(const int* __restrict__ dst, float* __restrict__ deg, int E) {
  int e = blockIdx.x * blockDim.x + threadIdx.x;
  if (e < E) atomicAdd(&deg[dst[e]], 1.0f);
}

__global__ void deg_to_dinv(float* __restrict__ deg, int n) {
  int i = blockIdx.x * blockDim.x + threadIdx.x;
  if (i < n) deg[i] = 1.0f / sqrtf(deg[i]);   // deg >= 1 always (self-loop)
}

// -------------------- pack W1 into WMMA B-fragment order, one float2 per lane
// B 4x16 fp32 fragment (ISA 7.12.2, symmetric to A): lane L holds
// col (L&15), K rows 2*(L>>4) and 2*(L>>4)+1 of the 4-row step.
// Bpack[q][L] = { W1[(4q + 2*(L>>4)    )*16 + (L&15)],
//                 W1[(4q + 2*(L>>4) + 1)*16 + (L&15)] }
__global__ void pack_w1(const float* __restrict__ W1, float* __restrict__ Bpack) {
  int t = blockIdx.x * blockDim.x + threadIdx.x;   // t in [0, KSTEPS*32)
  if (t >= KSTEPS * 32) return;
  int q = t >> 5;
  int L = t & 31;
  int m  = L & 15;
  int k0 = 4 * q + 2 * (L >> 4);
  Bpack[2 * t]     = W1[(size_t)k0 * NH + m];
  Bpack[2 * t + 1] = W1[(size_t)(k0 + 1) * NH + m];
}

// ------------------------------------------------- GEMM1: H1[N,16] = X @ W1
// One wave per 16-row tile; V_WMMA_F32_16X16X4_F32, 128 K-steps.
// Per step: one coalesced b64 for A (per-row K pair) + one coalesced b64 for B
// (pre-packed, wave-invariant -> L0 hit). Accumulator stays in v[0:7].
__global__ __launch_bounds__(256) void gemm1_wmma(const float* __restrict__ X,
                                                  const float* __restrict__ Bpack,
                                                  float* __restrict__ H1, int N) {
  const int lane = threadIdx.x & 31;
  const int wave = threadIdx.x >> 5;
  const int tile = blockIdx.x * 8 + wave;
  const int r0 = tile * 16;
  if (r0 >= N) return;                    // wave-uniform exit: EXEC all-1 at WMMA

  const int m  = lane & 15;
  const int hi = lane >> 4;
  const float* xrow = X + (size_t)(r0 + m) * NF + 2 * hi;  // consecutive K pair
  const float* bptr = Bpack + 2 * lane;                    // fragment stream

  v8f c = {};
  #pragma unroll 4
  for (int q = 0; q < KSTEPS; ++q) {
    v2f a, b;
    float2 av = *(const float2*)(xrow + 4 * q);        // X[r0+m][4q+2hi .. +1]
    float2 bv = *(const float2*)(bptr + 64 * q);       // packed W1 fragment
    a.x = av.x;  a.y = av.y;
    b.x = bv.x;  b.y = bv.y;
    c = __builtin_amdgcn_wmma_f32_16x16x4_f32(
        /*neg_a=*/false, a, /*neg_b=*/false, b,
        /*c_mod=*/(short)0, c, /*reuse_a=*/false, /*reuse_b=*/false);
  }

  // C/D 16x16 fp32: VGPR v, lane L -> row v + 8*(L>>4), col (L&15)
  float* hrow = H1 + (size_t)(r0 + 8 * hi) * NH + m;
  #pragma unroll
  for (int v = 0; v < 8; ++v) hrow[(size_t)v * NH] = c[v];
}

// ------------------------- scatter layer 1: agg1[d] += h1[s] * dinv[s]*dinv[d]
// 16 lanes per edge (feature j), edges E..E+N-1 are self-loops
__global__ void scatter1(const int* __restrict__ src, const int* __restrict__ dst,
                         const float* __restrict__ dinv, const float* __restrict__ H1,
                         float* __restrict__ agg, int E, int N) {
  long long t = (long long)blockIdx.x * blockDim.x + threadIdx.x;
  int e = (int)(t >> 4);
  int j = (int)(t & 15);
  if (e >= E + N) return;
  int s, d; float w;
  if (e < E) { s = src[e]; d = dst[e]; w = dinv[s] * dinv[d]; }
  else       { s = d = e - E; float di = dinv[s]; w = di * di; }
  atomicAdd(&agg[(size_t)d * NH + j], H1[(size_t)s * NH + j] * w);
}

// ---------------- GEMM2: H2[N,40] = relu(agg1 + b1) @ W2  (W2,b1 staged in LDS)
__global__ __launch_bounds__(256) void gemm2(const float* __restrict__ agg1,
                                             const float* __restrict__ b1,
                                             const float* __restrict__ W2,
                                             float* __restrict__ H2, int N) {
  __shared__ float sW2[NH * NC];
  __shared__ float sb1[NH];
  for (int i = threadIdx.x; i < NH * NC; i += blockDim.x) sW2[i] = W2[i];
  if (threadIdx.x < NH) sb1[threadIdx.x] = b1[threadIdx.x];
  __syncthreads();

  int row = blockIdx.x * blockDim.x + threadIdx.x;
  if (row >= N) return;
  const float* a = agg1 + (size_t)row * NH;
  float acc[NC];
  #pragma unroll
  for (int c = 0; c < NC; ++c) acc[c] = 0.0f;
  #pragma unroll
  for (int k = 0; k < NH; ++k) {
    float ak = a[k] + sb1[k];
    ak = ak > 0.0f ? ak : 0.0f;            // ReLU between layers
    #pragma unroll
    for (int c = 0; c < NC; ++c) acc[c] = fmaf(ak, sW2[k * NC + c], acc[c]);
  }
  float* o = H2 + (size_t)row * NC;
  #pragma unroll
  for (int c = 0; c < NC; ++c) o[c] = acc[c];
}

// ---------------- scatter layer 2: logits[d] += h2[s] * w   (40 lanes per edge)
__global__ __launch_bounds__(320) void scatter2(const int* __restrict__ src,
                                                const int* __restrict__ dst,
                                                const float* __restrict__ dinv,
                                                const float* __restrict__ H2,
                                                float* __restrict__ logits,
                                                int E, int N) {
  int e = blockIdx.x * 8 + threadIdx.x / NC;
  int j = threadIdx.x % NC;
  if (e >= E + N) return;
  int s, d; float w;
  if (e < E) { s = src[e]; d = dst[e]; w = dinv[s] * dinv[d]; }
  else       { s = d = e - E; float di = dinv[s]; w = di * di; }
  atomicAdd(&logits[(size_t)d * NC + j], H2[(size_t)s * NC + j] * w);
}

// -------- finalize: logits += b2 (in place), out = softmax(logits) per row
__global__ void finalize(float* __restrict__ out, float* __restrict__ logits,
                         const float* __restrict__ b2, int N) {
  int row = blockIdx.x * blockDim.x + threadIdx.x;
  if (row >= N) return;
  float* l = logits + (size_t)row * NC;
  float v[NC];
  float m = -INFINITY;
  #pragma unroll
  for (int c = 0; c < NC; ++c) { v[c] = l[c] + b2[c]; m = fmaxf(m, v[c]); }
  #pragma unroll
  for (int c = 0; c < NC; ++c) l[c] = v[c];        // write logits output
  float s = 0.0f;
  #pragma unroll
  for (int c = 0; c < NC; ++c) { v[c] = expf(v[c] - m); s += v[c]; }
  float inv = 1.0f / s;
  float* o = out + (size_t)row * NC;
  #pragma unroll
  for (int c = 0; c < NC; ++c) o[c] = v[c] * inv;  // softmax output
}

// --------------------------------------------------------------------- launch
extern "C" void kernel_launch(void* const* d_in, const int* in_sizes, int n_in,
                              void* d_out, int out_size, void* d_ws, size_t ws_size,
                              hipStream_t stream) {
  const float* X  = (const float*)d_in[0];
  const int*   ei = (const int*)d_in[1];   // edge_index flattened [2,E]
  const float* W1 = (const float*)d_in[3];
  const float* b1 = (const float*)d_in[4];
  const float* W2 = (const float*)d_in[5];
  const float* b2 = (const float*)d_in[6];

  const int N = in_sizes[0] / NF;          // 100000
  const int E = in_sizes[1] / 2;           // 3200000
  const int* src = ei;
  const int* dst = ei + E;

  // workspace (floats): dinv[N] | H1[16N] | agg1[16N] | H2[40N] | Bpack[8192]
  float* ws    = (float*)d_ws;
  float* dinv  = ws;
  float* H1    = ws + (size_t)N;
  float* agg1  = H1 + (size_t)16 * N;
  float* H2    = agg1 + (size_t)16 * N;
  float* Bpack = H2 + (size_t)40 * N;

  float* out    = (float*)d_out;            // softmax  [N*40]
  float* logits = out + (size_t)N * NC;     // logits   [N*40] (accumulated here)

  // 1) symmetric normalization
  fill_f32<<<(N + 255) / 256, 256, 0, stream>>>(dinv, 1.0f, N);
  deg_count<<<(E + 255) / 256, 256, 0, stream>>>(dst, dinv, E);
  deg_to_dinv<<<(N + 255) / 256, 256, 0, stream>>>(dinv, N);

  // 2) layer-1 linear via fp32 WMMA (W1 pre-packed into fragment order)
  pack_w1<<<(KSTEPS * 32 + 255) / 256, 256, 0, stream>>>(W1, Bpack);
  int tiles = (N + 15) / 16;
  gemm1_wmma<<<(tiles + 7) / 8, 256, 0, stream>>>(X, Bpack, H1, N);

  // 3) layer-1 normalized scatter-add
  fill_f32<<<(16 * N + 255) / 256, 256, 0, stream>>>(agg1, 0.0f, 16 * N);
  long long t1 = (long long)(E + N) * 16;
  scatter1<<<(unsigned)((t1 + 255) / 256), 256, 0, stream>>>(src, dst, dinv, H1, agg1, E, N);

  // 4) layer-2 linear (bias+ReLU fused on input)
  gemm2<<<(N + 255) / 256, 256, 0, stream>>>(agg1, b1, W2, H2, N);

  // 5) layer-2 scatter-add directly into d_out's logits half
  fill_f32<<<(N * NC + 255) / 256, 256, 0, stream>>>(logits, 0.0f, N * NC);
  scatter2<<<(E + N + 7) / 8, 320, 0, stream>>>(src, dst, dinv, H2, logits, E, N);

  // 6) bias + softmax
  finalize<<<(N + 255) / 256, 256, 0, stream>>>(out, logits, b2, N);
}